// Spatial3DAttentionLayer_83056077570345
// MI455X (gfx1250) — compile-verified
//
#include <hip/hip_runtime.h>
#include <math.h>

// ---------------- problem constants ----------------
#define N_INPUT   306
#define N_OUTPUT  270
#define KK2       81            // 9*9 contraction for Z*layout
#define N_DROPOUT 4
#define DROP_R2   (0.2f * 0.2f)
#define BATCH     64
#define TIME      2000

// Padded ASoftmax in workspace:
//   rows: M padded to 320 (5 blocks * 64), cols: K padded to 320 (20 chunks * 16)
#define SM_ROWS   320
#define SKP       320
#define KC        16            // K rows staged per LDS chunk
#define TBLK      128           // T columns per block
#define NCHUNK    (SKP / KC)    // 20

typedef __attribute__((ext_vector_type(2))) float v2f;
typedef __attribute__((ext_vector_type(8))) float v8f;

static __device__ __forceinline__ float waveReduceMax(float v) {
    #pragma unroll
    for (int off = 16; off > 0; off >>= 1)
        v = fmaxf(v, __shfl_xor(v, off, 32));
    return v;
}
static __device__ __forceinline__ float waveReduceSum(float v) {
    #pragma unroll
    for (int off = 16; off > 0; off >>= 1)
        v += __shfl_xor(v, off, 32);
    return v;
}

// ---------------------------------------------------------------------------
// Kernel 1: A[o,i] = sum_k Z[o,k]*layout[k,i]; dropout mask; row softmax.
// Writes zero-padded ASoftmax (SM_ROWS x SKP) into workspace.
// One block per row o (320 blocks; rows >= 270 write zeros). blockDim = 320.
// ---------------------------------------------------------------------------
__global__ __launch_bounds__(320)
void k_attn_softmax(const float* __restrict__ Z,        // (270, 81)
                    const float* __restrict__ layout,   // (81, 306)
                    const float* __restrict__ coords,   // (306, 2)
                    const float* __restrict__ dropc,    // (4, 2)
                    float* __restrict__ S)              // (SM_ROWS, SKP) padded
{
    const int o = blockIdx.x;       // 0..319
    const int i = threadIdx.x;      // 0..319
    const int lane = threadIdx.x & 31;
    const int wv   = threadIdx.x >> 5;   // 0..9

    __shared__ float zrow[KK2];
    __shared__ float red[16];

    if (o < N_OUTPUT && i < KK2) zrow[i] = Z[o * KK2 + i];
    __syncthreads();

    float val = -__builtin_inff();
    if (o < N_OUTPUT && i < N_INPUT) {
        float acc = 0.0f;
        #pragma unroll 3
        for (int k = 0; k < KK2; ++k)
            acc = fmaf(zrow[k], layout[k * N_INPUT + i], acc);
        const float cx = coords[i * 2 + 0];
        const float cy = coords[i * 2 + 1];
        bool dropped = false;
        #pragma unroll
        for (int d = 0; d < N_DROPOUT; ++d) {
            const float dx = cx - dropc[d * 2 + 0];
            const float dy = cy - dropc[d * 2 + 1];
            dropped = dropped || ((dx * dx + dy * dy) <= DROP_R2);
        }
        val = dropped ? -__builtin_inff() : acc;
    }

    // --- block max ---
    float m = waveReduceMax(val);
    if (lane == 0) red[wv] = m;
    __syncthreads();
    if (threadIdx.x < 32) {
        float t = (threadIdx.x < 10) ? red[threadIdx.x] : -__builtin_inff();
        t = waveReduceMax(t);
        if (threadIdx.x == 0) red[0] = t;
    }
    __syncthreads();
    m = red[0];

    // --- exp + block sum ---
    float e = 0.0f;
    if (o < N_OUTPUT && i < N_INPUT && val > -__builtin_inff())
        e = __expf(val - m);
    float s = waveReduceSum(e);
    __syncthreads();
    if (lane == 0) red[wv] = s;
    __syncthreads();
    if (threadIdx.x < 32) {
        float t = (threadIdx.x < 10) ? red[threadIdx.x] : 0.0f;
        t = waveReduceSum(t);
        if (threadIdx.x == 0) red[0] = t;
    }
    __syncthreads();
    const float invSum = 1.0f / red[0];

    float outv = 0.0f;
    if (o < N_OUTPUT && i < N_INPUT) outv = e * invSum;
    S[(size_t)o * SKP + i] = outv;   // cols >=306 and rows >=270 end up 0
}

// ---------------------------------------------------------------------------
// Kernel 2: out[b,o,t] = sum_i S[o,i] * x[b,i,t]  via V_WMMA_F32_16X16X4_F32.
// Block = 256 threads (8 waves): 4 M-waves x 2 T-waves.
// Block tile: M=64, T=128, K looped in LDS-staged chunks of 16 (double buffer,
// GLOBAL_LOAD_ASYNC_TO_LDS_B128 + s_wait_asynccnt). Wave tile: 16M x 64T.
// Grid: (ceil(2000/128)=16, 5 M-blocks, 64 batches).
// A-frag (16x4 f32): lane%16 = M row; lane/16 selects K pair {0,1}/{2,3}.
// B-frag (4x16 f32): lane%16 = T col; lane/16 selects K pair; v0=K+0, v1=K+1.
// C/D (16x16 f32):   lane%16 = T col; VGPR r = M row (lane/16)*8 + r.
// ---------------------------------------------------------------------------
__global__ __launch_bounds__(256)
void k_gemm_wmma(const float* __restrict__ S,   // (SM_ROWS, SKP) padded
                 const float* __restrict__ x,   // (BATCH, N_INPUT, TIME)
                 float* __restrict__ out)       // (BATCH, N_OUTPUT, TIME)
{
    __shared__ float xs[2][KC * TBLK];          // 2 x 8KB double buffer

    const int tid  = threadIdx.x;
    const int lane = tid & 31;
    const int wave = tid >> 5;                  // 0..7
    const int l16  = lane & 15;
    const int half = lane >> 4;                 // 0 or 1
    const int mw   = wave & 3;                  // M sub-tile within block
    const int tw   = wave >> 2;                 // T half within block

    const int tBlock = blockIdx.x * TBLK;
    const int mBase  = blockIdx.y * 64;
    const int b      = blockIdx.z;

    const float* xb   = x + (size_t)b * N_INPUT * TIME;
    const float* Srow = S + (size_t)(mBase + mw * 16 + l16) * SKP + half * 2;

    // Async-stage one K-chunk (16 rows x 128 t, f32) of x into LDS buffer.
    // Each thread moves 2x16B. Source addresses are clamped (k<=305,
    // t-chunk<=1996, both 16B-aligned); clamped garbage only reaches
    // zero-weighted (S pad) or discarded (t>=2000) slots.
#define ASYNC_ISSUE(cidx, bufSel)                                             \
    do {                                                                      \
        const int kb_ = (cidx) * KC;                                          \
        _Pragma("unroll")                                                     \
        for (int it_ = 0; it_ < 2; ++it_) {                                   \
            const int f_ = it_ * 1024 + tid * 4;   /* flat float index */     \
            int kG_ = kb_ + (f_ >> 7);                                        \
            kG_ = kG_ < (N_INPUT - 1) ? kG_ : (N_INPUT - 1);                  \
            int tG_ = tBlock + (f_ & 127);                                    \
            tG_ = tG_ < (TIME - 4) ? tG_ : (TIME - 4);                        \
            const float* g_ = xb + (size_t)kG_ * TIME + tG_;                  \
            unsigned ldsOff_ = (unsigned)(size_t)&xs[bufSel][f_];             \
            asm volatile("global_load_async_to_lds_b128 %0, %1, off"          \
                         :: "v"(ldsOff_), "v"(g_) : "memory");                \
        }                                                                     \
    } while (0)

    v8f acc0 = {}, acc1 = {}, acc2 = {}, acc3 = {};

    ASYNC_ISSUE(0, 0);

    for (int c = 0; c < NCHUNK; ++c) {
        if (c + 1 < NCHUNK) {
            ASYNC_ISSUE(c + 1, (c + 1) & 1);
            // 2 newest (chunk c+1) may remain outstanding; chunk c is done.
            asm volatile("s_wait_asynccnt 0x2" ::: "memory");
        } else {
            asm volatile("s_wait_asynccnt 0x0" ::: "memory");
        }
        __syncthreads();   // chunk c visible to all waves

        const float* sp  = Srow + c * KC;                 // A: global, L1-hot
        const float* bbp = &xs[c & 1][tw * 64 + l16];     // B: LDS
        #pragma unroll
        for (int ks = 0; ks < 4; ++ks) {
            v2f a;
            a.x = sp[ks * 4];
            a.y = sp[ks * 4 + 1];
            const float* bp = bbp + (ks * 4 + half * 2) * TBLK;
            v2f b0, b1, b2, b3;
            b0.x = bp[0];   b0.y = bp[TBLK + 0];
            b1.x = bp[16];  b1.y = bp[TBLK + 16];
            b2.x = bp[32];  b2.y = bp[TBLK + 32];
            b3.x = bp[48];  b3.y = bp[TBLK + 48];
            acc0 = __builtin_amdgcn_wmma_f32_16x16x4_f32(false, a, false, b0,
                                                         (short)0, acc0, false, false);
            acc1 = __builtin_amdgcn_wmma_f32_16x16x4_f32(false, a, false, b1,
                                                         (short)0, acc1, false, false);
            acc2 = __builtin_amdgcn_wmma_f32_16x16x4_f32(false, a, false, b2,
                                                         (short)0, acc2, false, false);
            acc3 = __builtin_amdgcn_wmma_f32_16x16x4_f32(false, a, false, b3,
                                                         (short)0, acc3, false, false);
        }
        __syncthreads();   // all reads of buf[c&1] done before it is re-filled
    }
#undef ASYNC_ISSUE

    // ---- guarded store epilogue (divergence after all WMMAs is fine) ----
    const int oBase = mBase + mw * 16 + half * 8;
    const int tb    = tBlock + tw * 64 + l16;
    float* ob = out + ((size_t)b * N_OUTPUT + oBase) * TIME;

#define STORE_TILE(ACC, TILE)                                                 \
    do {                                                                      \
        const int t_ = tb + (TILE) * 16;                                      \
        if (t_ < TIME) {                                                      \
            _Pragma("unroll")                                                 \
            for (int r_ = 0; r_ < 8; ++r_)                                    \
                if (oBase + r_ < N_OUTPUT)                                    \
                    ob[(size_t)r_ * TIME + t_] = (ACC)[r_];                   \
        }                                                                     \
    } while (0)

    STORE_TILE(acc0, 0);
    STORE_TILE(acc1, 1);
    STORE_TILE(acc2, 2);
    STORE_TILE(acc3, 3);
#undef STORE_TILE
}

// ---------------------------------------------------------------------------
extern "C" void kernel_launch(void* const* d_in, const int* in_sizes, int n_in,
                              void* d_out, int out_size, void* d_ws, size_t ws_size,
                              hipStream_t stream) {
    const float* x      = (const float*)d_in[0];  // (64, 306, 2000)
    const float* Z      = (const float*)d_in[1];  // (270, 9, 9, 1)
    const float* layout = (const float*)d_in[2];  // (1, 9, 9, 306)
    const float* coords = (const float*)d_in[3];  // (306, 2)
    const float* dropc  = (const float*)d_in[4];  // (4, 2)
    float* out = (float*)d_out;                   // (64, 270, 2000)
    float* S   = (float*)d_ws;                    // SM_ROWS * SKP floats

    (void)in_sizes; (void)n_in; (void)out_size; (void)ws_size;

    k_attn_softmax<<<dim3(SM_ROWS, 1, 1), dim3(320, 1, 1), 0, stream>>>(
        Z, layout, coords, dropc, S);

    const int tTiles = (TIME + TBLK - 1) / TBLK;      // 16
    const int mTiles = (SM_ROWS + 63) / 64;           // 5
    k_gemm_wmma<<<dim3(tTiles, mTiles, BATCH), dim3(256, 1, 1), 0, stream>>>(
        S, x, out);
}